// SparseConv_54631984005457
// MI455X (gfx1250) — compile-verified
//
#include <hip/hip_runtime.h>

#define HD 257                     // logical grid (Hd = Wd = 257)
#define BATCH 8
#define NPTS 32768
#define PTOT (BATCH * NPTS)        // 262144
#define CIN 16
#define COUT 64
#define KDIM (9 * CIN)             // 144
#define KPAD 146                   // padded LDS row stride: conflict-free, 8B aligned
#define HP 259                     // padded grid: 1-cell zero border each side
#define WP 259
#define SPAD (BATCH * HP * WP)     // 536648 padded cells

typedef float v2f __attribute__((ext_vector_type(2)));
typedef float v8f __attribute__((ext_vector_type(8)));

// ---------------- zero workspace (must re-zero every call) ----------------
__global__ void SparseConv_zero_kernel(float* __restrict__ vox, float* __restrict__ cnt) {
    int i = blockIdx.x * blockDim.x + threadIdx.x;
    if (i < SPAD) {
        float4 z = make_float4(0.f, 0.f, 0.f, 0.f);
        float4* v4 = reinterpret_cast<float4*>(vox) + (size_t)i * 4;
        v4[0] = z; v4[1] = z; v4[2] = z; v4[3] = z;
        cnt[i] = 0.f;
    }
}

// ---------------- scatter (segment sum via f32 atomics, interior cells) ----------------
__global__ void SparseConv_scatter_kernel(const float* __restrict__ xyzp,
                                          const float* __restrict__ feat,
                                          float* __restrict__ vox,
                                          float* __restrict__ cnt) {
    int p = blockIdx.x * blockDim.x + threadIdx.x;
    if (p >= PTOT) return;
    float4 pt = reinterpret_cast<const float4*>(xyzp)[p];
    int y = min(max((int)rintf(pt.y * 256.f), 0), 256);
    int x = min(max((int)rintf(pt.x * 256.f), 0), 256);
    int b = p >> 15;                       // p / NPTS
    size_t cell = ((size_t)b * HP + (y + 1)) * WP + (x + 1);
    float* dst = vox + cell * CIN;
    atomicAdd(dst + 0, pt.w);
    atomicAdd(dst + 1, 1.0f - pt.w);
    const float* fp = feat + (size_t)p * 14;
#pragma unroll
    for (int c = 0; c < 14; ++c) atomicAdd(dst + 2 + c, fp[c]);
    atomicAdd(cnt + cell, 1.0f);
}

// ---------------- normalize: vox = sum / max(cnt,1) (borders stay 0) ----------------
__global__ void SparseConv_norm_kernel(float* __restrict__ vox, const float* __restrict__ cnt) {
    int i = blockIdx.x * blockDim.x + threadIdx.x;
    if (i < SPAD) {
        float inv = 1.0f / fmaxf(cnt[i], 1.0f);
        float4* v4 = reinterpret_cast<float4*>(vox) + (size_t)i * 4;
#pragma unroll
        for (int q = 0; q < 4; ++q) {
            float4 v = v4[q];
            v.x *= inv; v.y *= inv; v.z *= inv; v.w *= inv;
            v4[q] = v;
        }
    }
}

// ------- gather + WMMA GEMM: per wave (32 pts x 144) @ (144 x 64) -------
__global__ __launch_bounds__(128) void SparseConv_gemm_kernel(
        const float* __restrict__ xyzp,
        const float* __restrict__ Wt,     // (9,16,64) row-major = (144,64)
        const float* __restrict__ bias,   // (64,)
        const float* __restrict__ vox,    // padded (SPAD,16), borders zero
        float* __restrict__ out) {        // (PTOT,64)
    __shared__ float Wlds[COUT * KPAD];   // transposed [n][j], padded: 37376 B
    __shared__ float blds[COUT];

    int tid = threadIdx.x;
    // stage W transposed: Wlds[n*KPAD + j] = Wt[j*64 + n] (coalesced global reads)
    for (int i = tid; i < KDIM * COUT; i += 128) {
        int j = i >> 6;          // 0..143
        int n = i & 63;          // 0..63
        Wlds[n * KPAD + j] = Wt[i];
    }
    if (tid < COUT) blds[tid] = bias[tid];
    __syncthreads();

    int wave = tid >> 5;
    int lane = tid & 31;
    int m    = lane & 15;                 // row (A) / col (B,C) index in tile
    int half = lane >> 4;                 // 0: K pair {0,1}; 1: K pair {2,3}
    int o    = half * 2;
    int base = blockIdx.x * 128 + wave * 32;

    // two M-tiles of 16 points; 32-bit byte offsets so gathers lower to
    // global_load_b64 (SGPR base + VGPR offset + imm), no flat path
    unsigned noff[2][9];
#pragma unroll
    for (int s = 0; s < 2; ++s) {
        int p = base + s * 16 + m;
        float4 pt = reinterpret_cast<const float4*>(xyzp)[p];
        int y = min(max((int)rintf(pt.y * 256.f), 0), 256);
        int x = min(max((int)rintf(pt.x * 256.f), 0), 256);
        int b = p >> 15;
        int center = (b * HP + (y + 1)) * WP + (x + 1);
#pragma unroll
        for (int dy = -1; dy <= 1; ++dy)
#pragma unroll
            for (int dx = -1; dx <= 1; ++dx)
                noff[s][(dy + 1) * 3 + (dx + 1)] =
                    (unsigned)(center + dy * WP + dx) * (CIN * 4);
    }

    v8f acc[2][4];
#pragma unroll
    for (int s = 0; s < 2; ++s)
#pragma unroll
        for (int t = 0; t < 4; ++t)
#pragma unroll
            for (int e = 0; e < 8; ++e) acc[s][t][e] = 0.0f;

    const float* wrow = &Wlds[m * KPAD];  // this lane's B column, contiguous in j
    const char*  vbase = reinterpret_cast<const char*>(vox);

    // fully unrolled K loop: 9 neighbors x 4 sub-steps of K=4; N: 4 tiles of 16
#pragma unroll
    for (int kn = 0; kn < 9; ++kn) {
#pragma unroll
        for (int kk = 0; kk < 4; ++kk) {
            int c = kk * 4 + o;           // even offset inside the 16-ch block
            int j = kn * 16 + c;          // global K index of first element
            // zero border replaces validity masking: direct load -> WMMA A operand
            v2f av0 = *reinterpret_cast<const v2f*>(vbase + noff[0][kn] + c * 4);
            v2f av1 = *reinterpret_cast<const v2f*>(vbase + noff[1][kn] + c * 4);
#pragma unroll
            for (int t = 0; t < 4; ++t) {
                v2f bv = *reinterpret_cast<const v2f*>(wrow + t * 16 * KPAD + j);
                acc[0][t] = __builtin_amdgcn_wmma_f32_16x16x4_f32(
                    false, av0, false, bv, (short)0, acc[0][t], false, false);
                acc[1][t] = __builtin_amdgcn_wmma_f32_16x16x4_f32(
                    false, av1, false, bv, (short)0, acc[1][t], false, false);
            }
        }
    }

    // store: VGPR r holds row (r + 8*half), col m of each 16x16 tile
#pragma unroll
    for (int s = 0; s < 2; ++s) {
        int rowbase = base + s * 16 + half * 8;
#pragma unroll
        for (int t = 0; t < 4; ++t) {
            float bb = blds[t * 16 + m];
#pragma unroll
            for (int r = 0; r < 8; ++r) {
                out[(size_t)(rowbase + r) * COUT + t * 16 + m] = acc[s][t][r] + bb;
            }
        }
    }
}

extern "C" void kernel_launch(void* const* d_in, const int* in_sizes, int n_in,
                              void* d_out, int out_size, void* d_ws, size_t ws_size,
                              hipStream_t stream) {
    const float* xyzp = (const float*)d_in[0];
    const float* feat = (const float*)d_in[1];
    const float* Wt   = (const float*)d_in[2];
    const float* bias = (const float*)d_in[3];
    float* out = (float*)d_out;

    float* vox = (float*)d_ws;                      // SPAD * 16 floats (34.3 MB)
    float* cnt = vox + (size_t)SPAD * CIN;          // SPAD floats (2.1 MB)

    SparseConv_zero_kernel<<<(SPAD + 255) / 256, 256, 0, stream>>>(vox, cnt);
    SparseConv_scatter_kernel<<<PTOT / 256, 256, 0, stream>>>(xyzp, feat, vox, cnt);
    SparseConv_norm_kernel<<<(SPAD + 255) / 256, 256, 0, stream>>>(vox, cnt);
    SparseConv_gemm_kernel<<<PTOT / 128, 128, 0, stream>>>(xyzp, Wt, bias, vox, out);
}